// PaiNN_481036337830
// MI455X (gfx1250) — compile-verified
//
#include <hip/hip_runtime.h>
#include <hip/hip_bf16.h>
#include <math.h>
#include <stdint.h>

// ---------------------------------------------------------------------------
// PaiNN for MI455X (gfx1250, wave32).
//  * fp32 WMMA (V_WMMA_F32_16X16X4_F32) for all node-side GEMMs
//  * weights staged LDS-resident via the Tensor Data Mover (tensor_load_to_lds,
//    TENSORcnt), with TDM padding to kill LDS bank conflicts on transposed B
//  * fused VALU+atomic edge message kernel for the memory-bound scatter
// ---------------------------------------------------------------------------

#define N_NODES_  20000
#define N_EDGES_  200000
#define F_        64
#define N_RBF_    20
#define L_        3
#define CUTOFF_   5.0f
#define PI_       3.14159265358979323846f

typedef __attribute__((ext_vector_type(2))) float v2f;
typedef __attribute__((ext_vector_type(8))) float v8f;
typedef uint32_t u32x4 __attribute__((ext_vector_type(4)));
typedef uint32_t u32x8 __attribute__((ext_vector_type(8)));

__device__ __forceinline__ void atomic_add_f32(float* p, float v) {
    __hip_atomic_fetch_add(p, v, __ATOMIC_RELAXED, __HIP_MEMORY_SCOPE_AGENT);
}

__device__ __forceinline__ float silu_f(float x) {
    return x * (1.0f / (1.0f + __expf(-x)));
}

// ---------------------------------------------------------------------------
// TDM stage: copy NE contiguous dwords at gptr into LDS at lds_byte_off.
// 1D tile: tensor_dim0 = tile_dim0 = NE (<= 65535), tile_dim1 = 1.
// pad_en: insert 1 DWORD of LDS padding every 64 DWORDs (interval code 5,
// amount code 0) -> transposed 64-wide weights land at row stride 65.
// Issued by one wave; completion via TENSORcnt.
// ---------------------------------------------------------------------------
__device__ __forceinline__ void tdm_load_1d(uint32_t lds_byte_off,
                                            const void* gptr, uint32_t ne_dw,
                                            int pad_en)
{
    const uint64_t ga = (uint64_t)(uintptr_t)gptr;
    u32x4 g0; u32x8 g1; u32x4 g2; u32x4 g3;
    // group0: count=1 | lds_addr | global_addr[56:0] | type=2
    g0[0] = 1u;
    g0[1] = lds_byte_off;
    g0[2] = (uint32_t)ga;
    g0[3] = (uint32_t)((ga >> 32) & 0x01FFFFFFull) | (2u << 30);
    // group1: data_size=4B, optional padding, dims for a 1-row tile
    g1[0] = (2u << 16) | (pad_en ? ((1u << 20) | (5u << 22)) : 0u);
    g1[1] = (ne_dw & 0xFFFFu) << 16;            // tensor_dim0 [15:0]
    g1[2] = (ne_dw >> 16) | (1u << 16);         // tensor_dim0 [31:16] | tensor_dim1=1
    g1[3] = ne_dw << 16;                        // tile_dim0
    g1[4] = 1u;                                 // tile_dim1=1, tile_dim2=0
    g1[5] = ne_dw;                              // tensor_dim0_stride [31:0]
    g1[6] = 0u;
    g1[7] = 0u;
    // group2/3: degenerate higher dims (tile_dim3/4 = 0 -> unused)
    g2[0] = 1u; g2[1] = 1u; g2[2] = 0u; g2[3] = 0u;
    g3[0] = 0u; g3[1] = 0u; g3[2] = 0u; g3[3] = 0u;
    asm volatile("tensor_load_to_lds %0, %1, %2, %3"
                 :
                 : "s"(g0), "s"(g1), "s"(g2), "s"(g3)
                 : "memory");
    __builtin_amdgcn_s_wait_tensorcnt(0);
}

// ---------------------------------------------------------------------------
// fp32 WMMA GEMM: C[m,n] = act( sum_k A[m,k]*B[k,n] + bias[n] )
// B (K x NCOLS dwords, contiguous) is staged into LDS once per block via TDM
// and indexed with LDS strides (bl_rs, bl_cs):
//   row-major weights:      bl_rs = NCOLS, bl_cs = 1   (pad_en = 0)
//   transposed 64x64 (U,V): bl_rs = 1,     bl_cs = 65  (pad_en = 1)
// A addressing is strided (a_rs, a_cs); A_UNIT=true -> b64 vector loads.
// One wave computes one 16x16 tile per iteration; K fully unrolled.
//   A-frag: lane m=lane&15, kh=lane>>4 -> K = 2kh, 2kh+1 (ISA 7.12.2)
//   B-frag: lane n=lane&15, kh=lane>>4 -> K = 2kh, 2kh+1
//   C/D:    elem r -> row r+8*kh, col lane&15
// ---------------------------------------------------------------------------
template <int K, int NCOLS, bool A_UNIT>
__global__ __launch_bounds__(256) void wmma_gemm_f32(
    const float* __restrict__ A, int a_rs, int a_cs,
    const float* __restrict__ B, int bl_rs, int bl_cs, int pad_en,
    const float* __restrict__ bias,
    float* __restrict__ C, int c_rs, int c_cs,
    int M, int act)
{
    constexpr int NE = K * NCOLS;
    __shared__ float Bs[NE + NE / 64];

    if (threadIdx.x < 32) {
        tdm_load_1d((uint32_t)(uintptr_t)(&Bs[0]), B, (uint32_t)NE, pad_en);
    }
    __syncthreads();

    const int lane   = threadIdx.x & 31;
    const int wave   = (blockIdx.x * blockDim.x + threadIdx.x) >> 5;
    const int nwaves = (gridDim.x * blockDim.x) >> 5;
    constexpr int tn_cnt = NCOLS >> 4;
    const int ntiles = (M >> 4) * tn_cnt;
    const int lm = lane & 15;
    const int kh = lane >> 4;

    for (int t = wave; t < ntiles; t += nwaves) {
        const int m0 = (t / tn_cnt) << 4;
        const int n0 = (t % tn_cnt) << 4;
        const int n  = n0 + lm;
        const float* arow = A + (size_t)(m0 + lm) * a_rs;
        v8f acc = {};
#pragma unroll
        for (int k0 = 0; k0 < K; k0 += 4) {
            v2f a, b;
            if constexpr (A_UNIT) {
                a = *(const v2f*)(arow + k0 + 2 * kh);       // global_load_b64
            } else {
                a.x = arow[(k0 + 2 * kh + 0) * a_cs];
                a.y = arow[(k0 + 2 * kh + 1) * a_cs];
            }
            b.x = Bs[(k0 + 2 * kh + 0) * bl_rs + n * bl_cs]; // ds_load
            b.y = Bs[(k0 + 2 * kh + 1) * bl_rs + n * bl_cs];
            acc = __builtin_amdgcn_wmma_f32_16x16x4_f32(
                false, a, false, b, (short)0, acc, false, false);
        }
        const float bv = bias ? bias[n] : 0.0f;
#pragma unroll
        for (int r = 0; r < 8; ++r) {
            const int m = m0 + r + (kh << 3);
            float val = acc[r] + bv;
            if (act) val = silu_f(val);
            C[(size_t)m * c_rs + (size_t)n * c_cs] = val;
        }
    }
}

// ---------------------------------------------------------------------------
// Per-edge geometry (layer-invariant): d, unit direction, RBF basis, cutoff.
// ---------------------------------------------------------------------------
__global__ __launch_bounds__(256) void edge_pre_kernel(
    const float* __restrict__ ea, float* __restrict__ dirv,
    float* __restrict__ rbf, float* __restrict__ fcut)
{
    const int e = blockIdx.x * blockDim.x + threadIdx.x;
    if (e >= N_EDGES_) return;
    const float x = ea[e * 3 + 0], y = ea[e * 3 + 1], z = ea[e * 3 + 2];
    const float d = sqrtf(x * x + y * y + z * z);
    const float inv = 1.0f / d;
    dirv[e * 3 + 0] = x * inv;
    dirv[e * 3 + 1] = y * inv;
    dirv[e * 3 + 2] = z * inv;
    const float pd = PI_ * d / CUTOFF_;
    fcut[e] = (d < CUTOFF_) ? 0.5f * (__cosf(pd) + 1.0f) : 0.0f;
#pragma unroll
    for (int n = 1; n <= N_RBF_; ++n)
        rbf[e * N_RBF_ + n - 1] = __sinf((float)n * pd) * inv;
}

// ---------------------------------------------------------------------------
// Fused edge message + segment-sum scatter. 64 threads own one edge
// (thread = feature channel f). Wr/br in LDS; native fp32 global atomics.
// ---------------------------------------------------------------------------
#define EDGES_PER_BLOCK 4
__global__ __launch_bounds__(256) void edge_msg_kernel(
    const int* __restrict__ eidx, const float* __restrict__ rbf,
    const float* __restrict__ fcut, const float* __restrict__ dirv,
    const float* __restrict__ phi, const float* __restrict__ v,
    const float* __restrict__ Wr, const float* __restrict__ br,
    float* __restrict__ ds_acc, float* __restrict__ dv_acc)
{
    __shared__ float WrS[N_RBF_ * 3 * F_];
    __shared__ float brS[3 * F_];
    for (int i = threadIdx.x; i < N_RBF_ * 3 * F_; i += blockDim.x) WrS[i] = Wr[i];
    for (int i = threadIdx.x; i < 3 * F_; i += blockDim.x) brS[i] = br[i];
    __syncthreads();

    const int f   = threadIdx.x & 63;
    const int grp = threadIdx.x >> 6;
    const int estep = gridDim.x * EDGES_PER_BLOCK;

    for (int e = blockIdx.x * EDGES_PER_BLOCK + grp; e < N_EDGES_; e += estep) {
        // hide next iteration's gather latency behind the filter math
        __builtin_prefetch(rbf + (size_t)(e + estep) * N_RBF_, 0, 0);

        const int src = eidx[e];
        const int dst = eidx[N_EDGES_ + e];
        const float fc = fcut[e];
        float wl = brS[f], wd = brS[F_ + f], wr = brS[2 * F_ + f];
        const float* re = rbf + (size_t)e * N_RBF_;
#pragma unroll
        for (int k = 0; k < N_RBF_; ++k) {
            const float r = re[k];
            wl += r * WrS[k * 192 + f];
            wd += r * WrS[k * 192 + F_ + f];
            wr += r * WrS[k * 192 + 2 * F_ + f];
        }
        wl *= fc; wd *= fc; wr *= fc;

        const float* ph = phi + (size_t)src * 192;
        const float xl = ph[f] * wl;
        const float xd = ph[F_ + f] * wd;
        const float xr = ph[2 * F_ + f] * wr;

        atomic_add_f32(&ds_acc[(size_t)dst * F_ + f], xd);

        const float* vs = v + (size_t)src * 192 + f * 3;
        float* dvp = dv_acc + (size_t)dst * 192 + f * 3;
        const float* de = dirv + (size_t)e * 3;
#pragma unroll
        for (int c = 0; c < 3; ++c)
            atomic_add_f32(&dvp[c], vs[c] * xl + xr * de[c]);
    }
}

// s = ds (+ s if addS); v += dv
__global__ __launch_bounds__(256) void msg_residual_kernel(
    float* __restrict__ s, float* __restrict__ v,
    const float* __restrict__ ds, const float* __restrict__ dv, int addS)
{
    const int i0 = blockIdx.x * blockDim.x + threadIdx.x;
    const int stride = gridDim.x * blockDim.x;
    for (int j = i0; j < N_NODES_ * F_; j += stride)
        s[j] = addS ? (s[j] + ds[j]) : ds[j];
    for (int j = i0; j < N_NODES_ * F_ * 3; j += stride)
        v[j] += dv[j];
}

// cat[:, :64] = s ; cat[:, 64:] = ||Vv||_c
__global__ __launch_bounds__(256) void cat_kernel(
    const float* __restrict__ s, const float* __restrict__ Vv,
    float* __restrict__ cat)
{
    const int i = blockIdx.x * blockDim.x + threadIdx.x;
    if (i >= N_NODES_ * F_) return;
    const int n = i >> 6, f = i & 63;
    cat[(size_t)n * 128 + f] = s[i];
    const float* vv = Vv + (size_t)n * 192 + f * 3;
    cat[(size_t)n * 128 + 64 + f] =
        sqrtf(vv[0] * vv[0] + vv[1] * vv[1] + vv[2] * vv[2]);
}

// s += a_sv * <Uv,Vv> + a_ss ; v += a_vv * Uv
__global__ __launch_bounds__(256) void upd_apply_kernel(
    float* __restrict__ s, float* __restrict__ v,
    const float* __restrict__ a, const float* __restrict__ Uv,
    const float* __restrict__ Vv)
{
    const int i = blockIdx.x * blockDim.x + threadIdx.x;
    if (i >= N_NODES_ * F_) return;
    const int n = i >> 6, f = i & 63;
    const float* ar = a + (size_t)n * 192;
    const float a_vv = ar[f], a_sv = ar[F_ + f], a_ss = ar[2 * F_ + f];
    const float* up = Uv + (size_t)n * 192 + f * 3;
    const float* wp = Vv + (size_t)n * 192 + f * 3;
    const float dot = up[0] * wp[0] + up[1] * wp[1] + up[2] * wp[2];
    s[i] += a_sv * dot + a_ss;
    float* vp = v + (size_t)n * 192 + f * 3;
    vp[0] += a_vv * up[0];
    vp[1] += a_vv * up[1];
    vp[2] += a_vv * up[2];
}

__global__ __launch_bounds__(256) void zero_kernel(float* __restrict__ p, int n)
{
    const int i0 = blockIdx.x * blockDim.x + threadIdx.x;
    const int stride = gridDim.x * blockDim.x;
    for (int j = i0; j < n; j += stride) p[j] = 0.0f;
}

// ---------------------------------------------------------------------------
// Host side
// ---------------------------------------------------------------------------
template <int K, int NCOLS, bool A_UNIT>
static void launch_gemm(const float* A, int ars, int acs,
                        const float* B, int bl_rs, int bl_cs, int pad_en,
                        const float* bias, float* C, int crs, int ccs,
                        int M, int act, hipStream_t st)
{
    const int ntiles = (M / 16) * (NCOLS / 16);
    int blocks = (ntiles + 7) / 8;      // 8 waves per 256-thread block
    if (blocks > 512) blocks = 512;     // amortize the TDM stage per block
    wmma_gemm_f32<K, NCOLS, A_UNIT><<<blocks, 256, 0, st>>>(
        A, ars, acs, B, bl_rs, bl_cs, pad_en, bias, C, crs, ccs, M, act);
}

extern "C" void kernel_launch(void* const* d_in, const int* in_sizes, int n_in,
                              void* d_out, int out_size, void* d_ws, size_t ws_size,
                              hipStream_t stream)
{
    const float* in_s  = (const float*)d_in[0];
    const float* in_v  = (const float*)d_in[1];
    const int*   eidx  = (const int*)d_in[2];
    const float* eattr = (const float*)d_in[3];
    const float* mW1 = (const float*)d_in[4];
    const float* mb1 = (const float*)d_in[5];
    const float* mW2 = (const float*)d_in[6];
    const float* mb2 = (const float*)d_in[7];
    const float* mWr = (const float*)d_in[8];
    const float* mbr = (const float*)d_in[9];
    const float* uU  = (const float*)d_in[10];
    const float* uV  = (const float*)d_in[11];
    const float* uW1 = (const float*)d_in[12];
    const float* ub1 = (const float*)d_in[13];
    const float* uW2 = (const float*)d_in[14];
    const float* ub2 = (const float*)d_in[15];

    // Output doubles as the live state: s (N*F) then v (N*F*3), return order.
    float* s_cur = (float*)d_out;
    float* v_cur = s_cur + (size_t)N_NODES_ * F_;

    hipMemcpyAsync(s_cur, in_s, (size_t)N_NODES_ * F_ * sizeof(float),
                   hipMemcpyDeviceToDevice, stream);
    hipMemcpyAsync(v_cur, in_v, (size_t)N_NODES_ * F_ * 3 * sizeof(float),
                   hipMemcpyDeviceToDevice, stream);

    // Workspace partition (~101 MB of fp32)
    float* ws     = (float*)d_ws;
    float* dirv   = ws;                                 // E*3
    float* rbf    = dirv + (size_t)N_EDGES_ * 3;        // E*20
    float* fcut   = rbf + (size_t)N_EDGES_ * N_RBF_;    // E
    float* ds_acc = fcut + N_EDGES_;                    // N*F
    float* dv_acc = ds_acc + (size_t)N_NODES_ * F_;     // N*F*3 (contig after ds)
    float* hbuf   = dv_acc + (size_t)N_NODES_ * F_ * 3; // N*F   (h, then g)
    float* nf3    = hbuf + (size_t)N_NODES_ * F_;       // N*3F  (phi, then a)
    float* Uvb    = nf3 + (size_t)N_NODES_ * F_ * 3;    // N*F*3
    float* Vvb    = Uvb + (size_t)N_NODES_ * F_ * 3;    // N*F*3
    float* catb   = Vvb + (size_t)N_NODES_ * F_ * 3;    // N*2F

    const int NF = N_NODES_ * F_;

    edge_pre_kernel<<<(N_EDGES_ + 255) / 256, 256, 0, stream>>>(eattr, dirv, rbf, fcut);

    for (int i = 0; i < L_; ++i) {
        // --- message block ---
        zero_kernel<<<2048, 256, 0, stream>>>(ds_acc, NF * 4); // ds_acc + dv_acc

        // h = silu(s @ mW1 + mb1)
        launch_gemm<64, 64, true>(s_cur, 64, 1, mW1 + (size_t)i * F_ * F_, 64, 1, 0,
                                  mb1 + (size_t)i * F_, hbuf, 64, 1,
                                  N_NODES_, 1, stream);
        // phi = h @ mW2 + mb2
        launch_gemm<64, 192, true>(hbuf, 64, 1, mW2 + (size_t)i * F_ * 3 * F_, 192, 1, 0,
                                   mb2 + (size_t)i * 192, nf3, 192, 1,
                                   N_NODES_, 0, stream);

        edge_msg_kernel<<<4096, 256, 0, stream>>>(
            eidx, rbf, fcut, dirv, nf3, v_cur,
            mWr + (size_t)i * N_RBF_ * 192, mbr + (size_t)i * 192,
            ds_acc, dv_acc);

        msg_residual_kernel<<<2048, 256, 0, stream>>>(s_cur, v_cur, ds_acc, dv_acc,
                                                      i > 0 ? 1 : 0);

        // --- update block ---
        // Uv[n,f,c] = sum_g U[f,g] v[n,g,c]; U staged transposed in LDS with
        // TDM padding (row stride 65) -> conflict-free ds_load.
        for (int c = 0; c < 3; ++c) {
            launch_gemm<64, 64, false>(v_cur + c, 192, 3, uU + (size_t)i * F_ * F_,
                                       1, 65, 1, nullptr, Uvb + c, 192, 3,
                                       N_NODES_, 0, stream);
            launch_gemm<64, 64, false>(v_cur + c, 192, 3, uV + (size_t)i * F_ * F_,
                                       1, 65, 1, nullptr, Vvb + c, 192, 3,
                                       N_NODES_, 0, stream);
        }

        cat_kernel<<<(NF + 255) / 256, 256, 0, stream>>>(s_cur, Vvb, catb);

        // g = silu(cat @ uW1 + ub1)
        launch_gemm<128, 64, true>(catb, 128, 1, uW1 + (size_t)i * 2 * F_ * F_, 64, 1, 0,
                                   ub1 + (size_t)i * F_, hbuf, 64, 1,
                                   N_NODES_, 1, stream);
        // a = g @ uW2 + ub2
        launch_gemm<64, 192, true>(hbuf, 64, 1, uW2 + (size_t)i * F_ * 3 * F_, 192, 1, 0,
                                   ub2 + (size_t)i * 192, nf3, 192, 1,
                                   N_NODES_, 0, stream);

        upd_apply_kernel<<<(NF + 255) / 256, 256, 0, stream>>>(s_cur, v_cur, nf3,
                                                               Uvb, Vvb);
    }
}